// NodeModel_84885733638244
// MI455X (gfx1250) — compile-verified
//
#include <hip/hip_runtime.h>
#include <hip/hip_bf16.h>
#include <stdint.h>

#define F          128
#define OUTF       384      // 2*F_NODE + F_EDGE
#define TILE_E     64       // edges per tile
#define TILES_PB   4        // tiles per block
#define EDGES_PB   (TILE_E * TILES_PB)
#define THREADS    256      // 8 wave32
// bytes per tile of edge_attr = TILE_E*F*4 = 32768 -> 2048 x b128 -> 8 per thread
#define CHUNKS_PER_THREAD 8

// ---- CDNA5 async global->LDS DMA (ASYNCcnt path, CDNA5 ISA) ----
// NT hint: edge_attr is a one-shot 327MB stream; keep it out of the 192MB L2 so
// the atomic accumulator (15.4MB) and x (5.1MB) stay resident.
__device__ __forceinline__ void async_b128_nt(uint32_t lds_off, uint64_t sbase, uint32_t voff) {
  // GVS mode: LDS address from vdst; global address = SGPR base + 32-bit VGPR offset
  asm volatile("global_load_async_to_lds_b128 %0, %1, %2 th:TH_LOAD_NT"
               :: "v"(lds_off), "v"(voff), "s"(sbase)
               : "memory");
}
__device__ __forceinline__ void wait_async_le8() {
#if defined(__has_builtin) && __has_builtin(__builtin_amdgcn_s_wait_asynccnt)
  __builtin_amdgcn_s_wait_asynccnt(8);
#else
  asm volatile("s_wait_asynccnt 8" ::: "memory");
#endif
}
__device__ __forceinline__ void wait_async_0() {
#if defined(__has_builtin) && __has_builtin(__builtin_amdgcn_s_wait_asynccnt)
  __builtin_amdgcn_s_wait_asynccnt(0);
#else
  asm volatile("s_wait_asynccnt 0" ::: "memory");
#endif
}

// ---- zero output + counts with b128 stores (harness poisons buffers with 0xAA) ----
__global__ void zero_kernel(float4* __restrict__ out4, float4* __restrict__ cnt4,
                            int out_n4, int cnt_n4) {
  int i = blockIdx.x * blockDim.x + threadIdx.x;
  float4 z = make_float4(0.f, 0.f, 0.f, 0.f);
  if (i < out_n4) out4[i] = z;
  if (i < cnt_n4) cnt4[i] = z;
}

// ---- scatter-sum: async-stream edge_attr into LDS; float4 lanes; fp32 atomics into L2 ----
__global__ void __launch_bounds__(THREADS)
scatter_kernel(const float* __restrict__ x, const int* __restrict__ eidx,
               const float* __restrict__ eattr, float* __restrict__ out,
               float* __restrict__ cnt, int n_edges) {
  __shared__ __align__(16) float abuf[2][TILE_E * F];  // 64KB double-buffered attr tiles
  __shared__ int2 rcbuf[TILE_E];                       // {row*OUTF, col*F} per edge

  const int tid  = threadIdx.x;
  const int lane = tid & 31;
  const int wv   = tid >> 5;   // 0..7
  const int esel = wv >> 1;    // wave pair -> edge select 0..3
  const int half = wv & 1;     // 0: x[col] half, 1: edge_attr half
  const int block_base = blockIdx.x * EDGES_PB;
  const uint64_t attr_base = (uint64_t)(uintptr_t)eattr;
  const uint32_t max_byte = (uint32_t)n_edges * (F * 4u) - 16u;

  // one tile's DMA: 8 x b128 per thread, uniform per wave -> ASYNCcnt +8
  auto issue_tile = [&](int tb, int buf) {
    uint32_t lds0 = (uint32_t)(uintptr_t)(&abuf[buf][0]);
    uint32_t g0 = (uint32_t)tb * (F * 4u);
#pragma unroll
    for (int j = 0; j < CHUNKS_PER_THREAD; ++j) {
      uint32_t chunk = (uint32_t)(j * THREADS + tid) * 16u;
      uint32_t go = g0 + chunk;
      if (go > max_byte) go = max_byte;   // clamp tail in-bounds (dup load, unused)
      async_b128_nt(lds0 + chunk, attr_base, go);
    }
  };

  int cur = 0;
  issue_tile(block_base, 0);              // prologue: tile 0 in flight

#pragma unroll 1
  for (int t = 0; t < TILES_PB; ++t) {
    const int tb = block_base + t * TILE_E;

    // stage indices: precompute row/col bases; per-edge count atomic lives here,
    // out of the hot loop
    if (tid < TILE_E) {
      int e = tb + tid;
      if (e < n_edges) {
        int r = eidx[e];              // destination (scatter)
        int c = eidx[n_edges + e];    // source (gather)
        rcbuf[tid] = make_int2(r * OUTF, c * F);
        unsafeAtomicAdd(cnt + r, 1.0f);
      }
    }

    if (t + 1 < TILES_PB) {
      issue_tile(tb + TILE_E, cur ^ 1);   // prefetch next tile
      wait_async_le8();                   // in-order completion => tile t landed
    } else {
      wait_async_0();
    }
    __syncthreads();                      // all waves' DMA + index stores visible

    const float* ab = &abuf[cur][0];
#pragma unroll 4
    for (int it = 0; it < TILE_E / 4; ++it) {
      int e_local = it * 4 + esel;        // 4 edges in flight per iteration
      int eg = tb + e_local;
      if (eg < n_edges) {
        int2 rc = rcbuf[e_local];         // single ds_load_b64 broadcast
        float4 v;
        float* dst;
        if (half == 0) {                  // even waves: gather x[col] row (L2-resident)
          v = ((const float4*)(x + rc.y))[lane];
          dst = out + rc.x + F + lane * 4;
        } else {                          // odd waves: edge_attr from LDS (b128)
          v = ((const float4*)(ab + e_local * F))[lane];
          dst = out + rc.x + 2 * F + lane * 4;
        }
        unsafeAtomicAdd(dst + 0, v.x);    // global_atomic_add_f32, no-return
        unsafeAtomicAdd(dst + 1, v.y);
        unsafeAtomicAdd(dst + 2, v.z);
        unsafeAtomicAdd(dst + 3, v.w);
      }
    }
    __syncthreads();                      // done reading abuf[cur]/rcbuf before reuse
    cur ^= 1;
  }
}

// ---- finalize: out[:,0:128] = x ; out[:,128:384] /= max(count,1) ----
__global__ void finalize_kernel(const float4* __restrict__ x4,
                                const float* __restrict__ cnt,
                                float4* __restrict__ out4, int n_nodes) {
  int i = blockIdx.x * blockDim.x + threadIdx.x;
  int total = n_nodes * (OUTF / 4);       // 96 float4 per node
  if (i >= total) return;
  int n = i / (OUTF / 4);
  int q = i - n * (OUTF / 4);
  if (q < F / 4) {
    out4[i] = x4[n * (F / 4) + q];
  } else {
    float c = cnt[n];
    c = (c < 1.0f) ? 1.0f : c;
    float4 v = out4[i];
    v.x = v.x / c;                        // exact f32 division, matches reference
    v.y = v.y / c;
    v.z = v.z / c;
    v.w = v.w / c;
    out4[i] = v;
  }
}

extern "C" void kernel_launch(void* const* d_in, const int* in_sizes, int n_in,
                              void* d_out, int out_size, void* d_ws, size_t ws_size,
                              hipStream_t stream) {
  const float* x     = (const float*)d_in[0];   // [N, 128]
  const int*   eidx  = (const int*)d_in[1];     // [2, E]
  const float* eattr = (const float*)d_in[2];   // [E, 128]
  // d_in[3] = u, d_in[4] = batch : unused by the output
  float* out = (float*)d_out;                   // [N, 384]
  float* cnt = (float*)d_ws;                    // [N] scratch counts

  const int n_nodes = in_sizes[0] / F;
  const int n_edges = in_sizes[1] / 2;
  const int out_elems = n_nodes * OUTF;

  const int out_n4 = out_elems / 4;             // OUTF divisible by 4
  const int cnt_n4 = (n_nodes + 3) / 4;
  const int zmax = (out_n4 > cnt_n4) ? out_n4 : cnt_n4;
  zero_kernel<<<(zmax + 255) / 256, 256, 0, stream>>>(
      (float4*)out, (float4*)cnt, out_n4, cnt_n4);

  const int blocks = (n_edges + EDGES_PB - 1) / EDGES_PB;
  scatter_kernel<<<blocks, THREADS, 0, stream>>>(x, eidx, eattr, out, cnt, n_edges);

  const int total4 = n_nodes * (OUTF / 4);
  finalize_kernel<<<(total4 + 255) / 256, 256, 0, stream>>>(
      (const float4*)x, cnt, (float4*)out, n_nodes);
}